// KNNClassifier_25116968747365
// MI455X (gfx1250) — compile-verified
//
#include <hip/hip_runtime.h>
#include <hip/hip_bf16.h>

// KNN classifier for MI455X (gfx1250, wave32).
// score = ||t||^2 - 2*(X @ X_train^T) via v_wmma_f32_16x16x4_f32 (per-query
// ||x||^2 is rank-invariant, dropped). Each block: 32 queries (two WMMA M
// tiles sharing every B fragment -> 16 FLOP per L2 byte), one 4-way slice of
// the train set. Streaming top-8 with producer-side threshold filtering into
// per-query LDS candidate queues; slice results lex-merged + voted in pass 2.

typedef float v2f __attribute__((ext_vector_type(2)));
typedef float v8f __attribute__((ext_vector_type(8)));

#define DIMS     512
#define NTRAIN   100000
#define NQ       2048
#define KNN      8
#define QT       32            // queries per block = two WMMA M tiles
#define CHUNK    128           // train points per chunk = 8 waves * 16
#define SPLITS   4             // train-dim split (grid.y) when ws available
#define XSTRIDE  514           // LDS row stride (floats): 8B-aligned b64, bank-spread
#define SSTRIDE  132           // LDS score row stride: producer stores conflict-free
#define CAP      32            // candidate queue capacity per query per chunk
#define FLT_BIG  3.0e38f

__device__ __forceinline__ void topk_insert(float (&bd)[KNN], int (&bi)[KNN],
                                            float s, int n) {
    // lexicographic (score, index): matches top_k first-occurrence semantics
    if (s < bd[KNN - 1] || (s == bd[KNN - 1] && n < bi[KNN - 1])) {
        bd[KNN - 1] = s; bi[KNN - 1] = n;
#pragma unroll
        for (int p = KNN - 1; p >= 1; --p) {
            if (bd[p] < bd[p - 1] || (bd[p] == bd[p - 1] && bi[p] < bi[p - 1])) {
                const float td = bd[p]; bd[p] = bd[p - 1]; bd[p - 1] = td;
                const int   ti = bi[p]; bi[p] = bi[p - 1]; bi[p - 1] = ti;
            }
        }
    }
}

__device__ __forceinline__ int vote8(const int (&labels)[KNN]) {
    int bestc = -1, bestl = 0x7fffffff;
#pragma unroll
    for (int j = 0; j < KNN; ++j) {
        int c = 0;
#pragma unroll
        for (int m = 0; m < KNN; ++m) c += (labels[m] == labels[j]) ? 1 : 0;
        if (c > bestc || (c == bestc && labels[j] < bestl)) { bestc = c; bestl = labels[j]; }
    }
    return bestl;
}

template <bool USE_WS>
__global__ __launch_bounds__(256) void knn_gemm_topk_kernel(
    const float* __restrict__ X,        // [NQ, DIMS]
    const float* __restrict__ Xt,       // [NTRAIN, DIMS]
    const int*   __restrict__ ytrain,   // [NTRAIN]
    int*         __restrict__ out,      // [NQ]
    float*       __restrict__ wsD,      // [nslices, NQ, KNN] scores
    int*         __restrict__ wsI,      // [nslices, NQ, KNN] indices
    int nslices)
{
    __shared__ float ldsX[QT * XSTRIDE];   // 32x512 query tile, padded
    __shared__ float ldsS[QT * SSTRIDE];   // per-chunk scores (overflow fallback)
    __shared__ float ldsThr[QT];           // per-query current 8th-best
    __shared__ int   ldsCnt[QT];           // per-query candidate count
    __shared__ float candS[QT * CAP];
    __shared__ int   candI[QT * CAP];

    const int tid     = threadIdx.x;
    const int lane    = tid & 31;
    const int wave    = tid >> 5;
    const int qbase   = blockIdx.x * QT;
    const int l15     = lane & 15;
    const int halfsel = lane >> 4;         // 0: K%4 in {0,1}; 1: K%4 in {2,3}

    const int slice_len  = (NTRAIN + nslices - 1) / nslices;
    const int slice_base = blockIdx.y * slice_len;
    const int slice_end  = (slice_base + slice_len < NTRAIN) ? (slice_base + slice_len)
                                                             : NTRAIN;

    // ---- cooperative load of 32x512 query tile into LDS + state init ----
    for (int i = tid; i < QT * DIMS; i += 256) {
        const int r = i >> 9;              // /512
        const int c = i & 511;
        ldsX[r * XSTRIDE + c] = X[(size_t)(qbase + r) * DIMS + c];
    }
    if (tid < QT) { ldsThr[tid] = FLT_BIG; ldsCnt[tid] = 0; }
    __syncthreads();

    // ---- per-owner top-8 (thread t owns query qbase+t, t<32), sorted asc ----
    float bd[KNN];
    int   bi[KNN];
#pragma unroll
    for (int j = 0; j < KNN; ++j) { bd[j] = FLT_BIG; bi[j] = 0; }

    // A fragments: tile0 = queries 0..15 (LDS rows 0..15), tile1 = 16..31
    const float* aptr0 = &ldsX[l15 * XSTRIDE + 2 * halfsel];
    const float* aptr1 = aptr0 + 16 * XSTRIDE;

    const int nchunks = (slice_end - slice_base + CHUNK - 1) / CHUNK;
    for (int ch = 0; ch < nchunks; ++ch) {
        const int  tb    = slice_base + ch * CHUNK;
        const int  ncol  = tb + wave * 16 + l15;        // global train index
        const bool valid = (ncol < slice_end);
        const int  trow  = valid ? ncol : (slice_end - 1); // clamp: EXEC all-1s for WMMA
        const float* bptr = Xt + (size_t)trow * DIMS + 2 * halfsel;

        // prefetch next chunk's row for this lane (global_prefetch_b8 path)
        {
            const int pn = (ncol + CHUNK < slice_end) ? (ncol + CHUNK) : (slice_end - 1);
            __builtin_prefetch(Xt + (size_t)pn * DIMS + 2 * halfsel, 0, 1);
        }

        // K loop: 256 WMMA per chunk; each B fragment feeds both A tiles.
        // Four accumulator chains (2 tiles x even/odd k) + 2 fmac chains.
        v8f  accA0 = {}, accA1 = {}, accB0 = {}, accB1 = {};
        float t2a = 0.0f, t2b = 0.0f;
#pragma unroll 4
        for (int k0 = 0; k0 < DIMS; k0 += 4) {
            const v2f a0 = *(const v2f*)(aptr0 + k0);   // ds_load_b64
            const v2f a1 = *(const v2f*)(aptr1 + k0);   // ds_load_b64
            const v2f b  = *(const v2f*)(bptr  + k0);   // global_load_b64 (shared)
            if ((k0 >> 2) & 1) {
                t2b = __builtin_fmaf(b.x, b.x, t2b);
                t2b = __builtin_fmaf(b.y, b.y, t2b);
                accA1 = __builtin_amdgcn_wmma_f32_16x16x4_f32(
                            false, a0, false, b, (short)0, accA1, false, false);
                accB1 = __builtin_amdgcn_wmma_f32_16x16x4_f32(
                            false, a1, false, b, (short)0, accB1, false, false);
            } else {
                t2a = __builtin_fmaf(b.x, b.x, t2a);
                t2a = __builtin_fmaf(b.y, b.y, t2a);
                accA0 = __builtin_amdgcn_wmma_f32_16x16x4_f32(
                            false, a0, false, b, (short)0, accA0, false, false);
                accB0 = __builtin_amdgcn_wmma_f32_16x16x4_f32(
                            false, a1, false, b, (short)0, accB0, false, false);
            }
        }
        // combine the two K-halves of ||t||^2 (lanes n and n+16 share column n)
        const float t2p = t2a + t2b;
        const float t2  = t2p + __shfl_xor(t2p, 16, 32);

        // emit scores: C VGPR i -> (row = i + 8*halfsel [+16 for tile1], col = l15)
        const int col = wave * 16 + l15;
#pragma unroll
        for (int i = 0; i < 8; ++i) {
            const int   rowA = i + 8 * halfsel;
            const int   rowB = rowA + 16;
            const float sA = valid ? (t2 - 2.0f * (accA0[i] + accA1[i])) : FLT_BIG;
            const float sB = valid ? (t2 - 2.0f * (accB0[i] + accB1[i])) : FLT_BIG;
            ldsS[rowA * SSTRIDE + col] = sA;
            ldsS[rowB * SSTRIDE + col] = sB;
            if (valid && sA <= ldsThr[rowA]) {          // <= : keep equal-score ties
                const int pos = atomicAdd(&ldsCnt[rowA], 1);    // ds_add_rtn_u32
                if (pos < CAP) { candS[rowA * CAP + pos] = sA; candI[rowA * CAP + pos] = ncol; }
            }
            if (valid && sB <= ldsThr[rowB]) {
                const int pos = atomicAdd(&ldsCnt[rowB], 1);
                if (pos < CAP) { candS[rowB * CAP + pos] = sB; candI[rowB * CAP + pos] = ncol; }
            }
        }
        __syncthreads();

        // owner lanes merge candidates (full row scan only on queue overflow,
        // i.e. essentially just the first chunk while thresholds are +INF)
        if (tid < QT) {
            const int cnt = ldsCnt[tid];
            if (cnt > CAP) {
                const float* srow = &ldsS[tid * SSTRIDE];
                for (int j = 0; j < CHUNK; ++j)
                    topk_insert(bd, bi, srow[j], tb + j);
            } else {
                for (int j = 0; j < cnt; ++j)
                    topk_insert(bd, bi, candS[tid * CAP + j], candI[tid * CAP + j]);
            }
            ldsThr[tid] = bd[KNN - 1];   // publish new threshold
            ldsCnt[tid] = 0;             // reset queue
        }
        __syncthreads();
    }

    // ---- emit: per-slice candidates to workspace, or vote directly ----
    if (tid < QT) {
        if (USE_WS) {
            const size_t o = ((size_t)blockIdx.y * NQ + (qbase + tid)) * KNN;
#pragma unroll
            for (int j = 0; j < KNN; ++j) { wsD[o + j] = bd[j]; wsI[o + j] = bi[j]; }
        } else {
            int labels[KNN];
#pragma unroll
            for (int j = 0; j < KNN; ++j) labels[j] = ytrain[bi[j]];
            out[qbase + tid] = vote8(labels);
        }
    }
}

__global__ __launch_bounds__(256) void knn_merge_vote_kernel(
    const float* __restrict__ wsD, const int* __restrict__ wsI,
    const int* __restrict__ ytrain, int* __restrict__ out, int nslices)
{
    const int q = blockIdx.x * blockDim.x + threadIdx.x;
    if (q >= NQ) return;

    float bd[KNN];
    int   bi[KNN];
#pragma unroll
    for (int j = 0; j < KNN; ++j) { bd[j] = FLT_BIG; bi[j] = 0; }

    for (int s = 0; s < nslices; ++s) {
        const size_t o = ((size_t)s * NQ + q) * KNN;
#pragma unroll
        for (int j = 0; j < KNN; ++j)
            topk_insert(bd, bi, wsD[o + j], wsI[o + j]);
    }

    int labels[KNN];
#pragma unroll
    for (int j = 0; j < KNN; ++j) labels[j] = ytrain[bi[j]];
    out[q] = vote8(labels);
}

extern "C" void kernel_launch(void* const* d_in, const int* in_sizes, int n_in,
                              void* d_out, int out_size, void* d_ws, size_t ws_size,
                              hipStream_t stream) {
    const float* X      = (const float*)d_in[0];   // [2048, 512]
    const float* Xtrain = (const float*)d_in[1];   // [100000, 512]
    const int*   ytrain = (const int*)d_in[2];     // [100000]
    // d_in[3] is k (== 8), baked in as KNN at compile time
    int* out = (int*)d_out;                        // [2048] int32 predictions

    const size_t need = (size_t)SPLITS * NQ * KNN * (sizeof(float) + sizeof(int));
    if (d_ws != nullptr && ws_size >= need) {
        float* wsD = (float*)d_ws;
        int*   wsI = (int*)(wsD + (size_t)SPLITS * NQ * KNN);
        dim3 grid(NQ / QT, SPLITS);   // 64 x 4 = 256 blocks
        hipLaunchKernelGGL((knn_gemm_topk_kernel<true>), grid, dim3(256), 0, stream,
                           X, Xtrain, ytrain, out, wsD, wsI, SPLITS);
        hipLaunchKernelGGL(knn_merge_vote_kernel, dim3(NQ / 256), dim3(256), 0, stream,
                           wsD, wsI, ytrain, out, SPLITS);
    } else {
        // deterministic fallback: single slice, vote inline, no workspace
        dim3 grid(NQ / QT, 1);
        hipLaunchKernelGGL((knn_gemm_topk_kernel<false>), grid, dim3(256), 0, stream,
                           X, Xtrain, ytrain, out, nullptr, nullptr, 1);
    }
}